// GATGRUCell_42958262894962
// MI455X (gfx1250) — compile-verified
//
#include <hip/hip_runtime.h>
#include <hip/hip_bf16.h>
#include <math.h>

typedef __attribute__((ext_vector_type(2))) float v2f;
typedef __attribute__((ext_vector_type(8))) float v8f;

#define DIM 128
#define LSTR 132   // LDS row stride (floats): 132 % 64 == 4 -> conflict-free column access
#define ALPHA 0.2f

__device__ __forceinline__ v8f wmma_f32(v2f a, v2f b, v8f c) {
    // V_WMMA_F32_16X16X4_F32 : D = A(16x4) * B(4x16) + C(16x16), all fp32
    return __builtin_amdgcn_wmma_f32_16x16x4_f32(false, a, false, b, (short)0, c, false, false);
}

__device__ __forceinline__ void atomicMaxF(float* addr, float val) {
    int* ai = (int*)addr;
    int old = __float_as_int(*addr);
    while (val > __int_as_float(old)) {
        int assumed = old;
        old = atomicCAS(ai, assumed, __float_as_int(val));
        if (old == assumed) break;
    }
}

// ---------------- init: zero h' accumulator, m=-inf, denom=0 ----------------
__global__ void init_kernel(float* __restrict__ hp, float* __restrict__ m,
                            float* __restrict__ denom, int N) {
    long total = (long)N * DIM;
    long stride = (long)gridDim.x * blockDim.x;
    long g = (long)blockIdx.x * blockDim.x + threadIdx.x;
    for (long i = g; i < total; i += stride) hp[i] = 0.f;
    for (long i = g; i < N; i += stride) { m[i] = -INFINITY; denom[i] = 0.f; }
}

// ---------------- Wh = h @ W^T + b_W  (WMMA f32) ----------------
// block: 256 thr = 8 waves; block owns 16 node rows; wave w owns cols [16w,16w+16)
__global__ void wh_kernel(const float* __restrict__ h, const float* __restrict__ W,
                          const float* __restrict__ bW, float* __restrict__ Wh) {
    __shared__ float As[16][LSTR];
    const int node0 = blockIdx.x * 16;
    const int tid = threadIdx.x;
    for (int idx = tid; idx < 16 * (DIM / 4); idx += 256) {
        int row = idx >> 5;              // /32
        int col = (idx & 31) * 4;
        float4 v = *(const float4*)(h + (size_t)(node0 + row) * DIM + col);
        As[row][col + 0] = v.x; As[row][col + 1] = v.y;
        As[row][col + 2] = v.z; As[row][col + 3] = v.w;
    }
    __syncthreads();
    const int wave = tid >> 5, lane = tid & 31;
    const int half = lane >> 4, l16 = lane & 15;
    const int f0 = wave * 16;
    const float* Brow = W + (size_t)(f0 + l16) * DIM;   // B[k][n] = W[n][k]
    v8f acc = {};
#pragma unroll 8
    for (int k0 = 0; k0 < DIM; k0 += 4) {
        int ks = k0 + half * 2;
        v2f a; a.x = As[l16][ks]; a.y = As[l16][ks + 1];
        v2f b; b.x = Brow[ks];    b.y = Brow[ks + 1];
        acc = wmma_f32(a, b, acc);
    }
    const float bias = bW[f0 + l16];
#pragma unroll
    for (int r = 0; r < 8; ++r) {
        int mrow = r + half * 8;
        Wh[(size_t)(node0 + mrow) * DIM + f0 + l16] = acc[r] + bias;
    }
}

// ---------------- s_s = Wh@a_src, s_d = Wh@a_dst (one wave per node) ----------------
__global__ void sdot_kernel(const float* __restrict__ Wh, const float* __restrict__ as,
                            const float* __restrict__ ad, float* __restrict__ ss,
                            float* __restrict__ sd, int N) {
    int wave = blockIdx.x * (blockDim.x >> 5) + (threadIdx.x >> 5);
    int lane = threadIdx.x & 31;
    if (wave >= N) return;
    float4 w = *(const float4*)(Wh + (size_t)wave * DIM + lane * 4);
    float4 s = *(const float4*)(as + lane * 4);
    float4 d = *(const float4*)(ad + lane * 4);
    float v1 = w.x * s.x + w.y * s.y + w.z * s.z + w.w * s.w;
    float v2 = w.x * d.x + w.y * d.y + w.z * d.z + w.w * d.w;
    for (int off = 16; off; off >>= 1) {
        v1 += __shfl_xor(v1, off, 32);
        v2 += __shfl_xor(v2, off, 32);
    }
    if (lane == 0) { ss[wave] = v1; sd[wave] = v2; }
}

// ---------------- edge leaky-relu + segment max ----------------
__global__ void edge_max_kernel(const int* __restrict__ src, const int* __restrict__ dst,
                                const float* __restrict__ ss, const float* __restrict__ sd,
                                const float* __restrict__ ab, float* __restrict__ ev,
                                float* __restrict__ m, int E) {
    int e = blockIdx.x * blockDim.x + threadIdx.x;
    if (e >= E) return;
    float v = ss[src[e]] + sd[dst[e]] + ab[0];
    v = v > 0.f ? v : ALPHA * v;
    ev[e] = v;
    atomicMaxF(&m[dst[e]], v);
}

// ---------------- edge exp + segment sum ----------------
__global__ void edge_exp_kernel(const int* __restrict__ dst, float* __restrict__ ev,
                                const float* __restrict__ m, float* __restrict__ denom, int E) {
    int e = blockIdx.x * blockDim.x + threadIdx.x;
    if (e >= E) return;
    int d = dst[e];
    float ex = expf(ev[e] - m[d]);
    ev[e] = ex;
    atomicAdd(&denom[d], ex);
}

// ---------------- weighted scatter-add: one wave per edge ----------------
__global__ void scatter_kernel(const int* __restrict__ src, const int* __restrict__ dst,
                               const float* __restrict__ ev, const float* __restrict__ denom,
                               const float* __restrict__ Wh, float* __restrict__ hp, int E) {
    int wave = blockIdx.x * (blockDim.x >> 5) + (threadIdx.x >> 5);
    int lane = threadIdx.x & 31;
    if (wave >= E) return;
    int s = src[wave], d = dst[wave];
    float den = denom[d];
    float attn = ev[wave] / (den > 0.f ? den : 1.f);
    float4 w = *(const float4*)(Wh + (size_t)s * DIM + lane * 4);
    float* o = hp + (size_t)d * DIM + lane * 4;
    atomicAdd(o + 0, attn * w.x);
    atomicAdd(o + 1, attn * w.y);
    atomicAdd(o + 2, attn * w.z);
    atomicAdd(o + 3, attn * w.w);
}

// ---------------- h' += h @ res_W^T + res_b (in place, WMMA f32) ----------------
__global__ void res_kernel(const float* __restrict__ h, const float* __restrict__ rW,
                           const float* __restrict__ rb, float* __restrict__ hp) {
    __shared__ float As[16][LSTR];
    const int node0 = blockIdx.x * 16;
    const int tid = threadIdx.x;
    for (int idx = tid; idx < 16 * (DIM / 4); idx += 256) {
        int row = idx >> 5;
        int col = (idx & 31) * 4;
        float4 v = *(const float4*)(h + (size_t)(node0 + row) * DIM + col);
        As[row][col + 0] = v.x; As[row][col + 1] = v.y;
        As[row][col + 2] = v.z; As[row][col + 3] = v.w;
    }
    __syncthreads();
    const int wave = tid >> 5, lane = tid & 31;
    const int half = lane >> 4, l16 = lane & 15;
    const int f0 = wave * 16;
    const float* Brow = rW + (size_t)(f0 + l16) * DIM;
    v8f acc = {};
#pragma unroll 8
    for (int k0 = 0; k0 < DIM; k0 += 4) {
        int ks = k0 + half * 2;
        v2f a; a.x = As[l16][ks]; a.y = As[l16][ks + 1];
        v2f b; b.x = Brow[ks];    b.y = Brow[ks + 1];
        acc = wmma_f32(a, b, acc);
    }
    const float bias = rb[f0 + l16];
#pragma unroll
    for (int r = 0; r < 8; ++r) {
        int mrow = r + half * 8;
        size_t off = (size_t)(node0 + mrow) * DIM + f0 + l16;
        hp[off] = hp[off] + acc[r] + bias;
    }
}

// ---------------- fused GRU cell: 6 WMMA-accumulated tiles + gates ----------------
__global__ void gru_kernel(const float* __restrict__ h, const float* __restrict__ hp,
                           const float* __restrict__ Wi, const float* __restrict__ Wg,
                           const float* __restrict__ bi, const float* __restrict__ bh,
                           float* __restrict__ out) {
    __shared__ float Hs[16][LSTR];   // hidden h strip
    __shared__ float Ps[16][LSTR];   // h' strip (GRU input)
    const int node0 = blockIdx.x * 16;
    const int tid = threadIdx.x;
    for (int idx = tid; idx < 16 * (DIM / 4); idx += 256) {
        int row = idx >> 5;
        int col = (idx & 31) * 4;
        float4 v = *(const float4*)(h  + (size_t)(node0 + row) * DIM + col);
        float4 p = *(const float4*)(hp + (size_t)(node0 + row) * DIM + col);
        Hs[row][col + 0] = v.x; Hs[row][col + 1] = v.y;
        Hs[row][col + 2] = v.z; Hs[row][col + 3] = v.w;
        Ps[row][col + 0] = p.x; Ps[row][col + 1] = p.y;
        Ps[row][col + 2] = p.z; Ps[row][col + 3] = p.w;
    }
    __syncthreads();
    const int wave = tid >> 5, lane = tid & 31;
    const int half = lane >> 4, l16 = lane & 15;
    const int f0 = wave * 16;
    const int f = f0 + l16;
    const float* Wir = Wi + (size_t)(f) * DIM;
    const float* Wiz = Wi + (size_t)(DIM + f) * DIM;
    const float* Win = Wi + (size_t)(2 * DIM + f) * DIM;
    const float* Wgr = Wg + (size_t)(f) * DIM;
    const float* Wgz = Wg + (size_t)(DIM + f) * DIM;
    const float* Wgn = Wg + (size_t)(2 * DIM + f) * DIM;
    v8f air = {}, aiz = {}, ain = {}, ahr = {}, ahz = {}, ahn = {};
#pragma unroll 4
    for (int k0 = 0; k0 < DIM; k0 += 4) {
        int ks = k0 + half * 2;
        v2f ah; ah.x = Hs[l16][ks]; ah.y = Hs[l16][ks + 1];
        v2f ap; ap.x = Ps[l16][ks]; ap.y = Ps[l16][ks + 1];
        v2f b;
        b.x = Wir[ks]; b.y = Wir[ks + 1]; air = wmma_f32(ap, b, air);
        b.x = Wiz[ks]; b.y = Wiz[ks + 1]; aiz = wmma_f32(ap, b, aiz);
        b.x = Win[ks]; b.y = Win[ks + 1]; ain = wmma_f32(ap, b, ain);
        b.x = Wgr[ks]; b.y = Wgr[ks + 1]; ahr = wmma_f32(ah, b, ahr);
        b.x = Wgz[ks]; b.y = Wgz[ks + 1]; ahz = wmma_f32(ah, b, ahz);
        b.x = Wgn[ks]; b.y = Wgn[ks + 1]; ahn = wmma_f32(ah, b, ahn);
    }
    const float bir = bi[f], biz = bi[DIM + f], bin = bi[2 * DIM + f];
    const float bhr = bh[f], bhz = bh[DIM + f], bhn = bh[2 * DIM + f];
#pragma unroll
    for (int r = 0; r < 8; ++r) {
        int mrow = r + half * 8;
        float ir = air[r] + bir, hr = ahr[r] + bhr;
        float iz = aiz[r] + biz, hz = ahz[r] + bhz;
        float in_ = ain[r] + bin, hn = ahn[r] + bhn;
        float rg = 1.f / (1.f + expf(-(ir + hr)));
        float z  = 1.f / (1.f + expf(-(iz + hz)));
        float nn = tanhf(in_ + rg * hn);
        float hv = Hs[mrow][f];
        out[(size_t)(node0 + mrow) * DIM + f] = (1.f - z) * nn + z * hv;
    }
}

extern "C" void kernel_launch(void* const* d_in, const int* in_sizes, int n_in,
                              void* d_out, int out_size, void* d_ws, size_t ws_size,
                              hipStream_t stream) {
    const float* h    = (const float*)d_in[0];
    const int*   src  = (const int*)  d_in[1];
    const int*   dst  = (const int*)  d_in[2];
    const float* W    = (const float*)d_in[3];
    const float* bW   = (const float*)d_in[4];
    const float* asrc = (const float*)d_in[5];
    const float* adst = (const float*)d_in[6];
    const float* ab   = (const float*)d_in[7];
    const float* rW   = (const float*)d_in[8];
    const float* rb   = (const float*)d_in[9];
    const float* Wi   = (const float*)d_in[10];
    const float* Wg   = (const float*)d_in[11];
    const float* bi   = (const float*)d_in[12];
    const float* bh   = (const float*)d_in[13];
    float* out = (float*)d_out;

    const int N = in_sizes[0] / DIM;
    const int E = in_sizes[1];
    const size_t ND = (size_t)N * DIM;

    float* ws    = (float*)d_ws;
    float* Wh    = ws;              // N*D
    float* hp    = ws + ND;         // N*D  (h' accumulator, then h'+res in place)
    float* ss    = ws + 2 * ND;     // N
    float* sd    = ss + N;          // N
    float* m     = sd + N;          // N
    float* denom = m + N;           // N
    float* ev    = denom + N;       // E    (edge score, then exp in place)

    init_kernel<<<512, 256, 0, stream>>>(hp, m, denom, N);
    wh_kernel<<<N / 16, 256, 0, stream>>>(h, W, bW, Wh);
    sdot_kernel<<<(N + 7) / 8, 256, 0, stream>>>(Wh, asrc, adst, ss, sd, N);
    edge_max_kernel<<<(E + 255) / 256, 256, 0, stream>>>(src, dst, ss, sd, ab, ev, m, E);
    edge_exp_kernel<<<(E + 255) / 256, 256, 0, stream>>>(dst, ev, m, denom, E);
    scatter_kernel<<<(E + 7) / 8, 256, 0, stream>>>(src, dst, ev, denom, Wh, hp, E);
    res_kernel<<<N / 16, 256, 0, stream>>>(h, rW, rb, hp);
    gru_kernel<<<N / 16, 256, 0, stream>>>(h, hp, Wi, Wg, bi, bh, out);
}